// SimpleMessagePassing_7876970021340
// MI455X (gfx1250) — compile-verified
//
#include <hip/hip_runtime.h>

typedef float v2f __attribute__((ext_vector_type(2)));
typedef float v8f __attribute__((ext_vector_type(8)));

constexpr int Bc = 4;           // batches
constexpr int Nc = 50000;       // points per batch
constexpr int Kc = 16;          // neighbors
constexpr int Dc = 128;         // feature dim
constexpr int TILE_M = 64;      // points per block (4 WMMA M-subtiles)
constexpr int SUBT   = TILE_M / 16;     // 4 subtiles
constexpr int LDS_STRIDE = 132; // padded fp32 row stride -> conflict-free A reads
constexpr int THREADS = 256;    // 8 wave32s

__global__ __launch_bounds__(THREADS)
void mp_pool_linear_kernel(const float* __restrict__ x,
                           const int*  __restrict__ knn,
                           const float* __restrict__ W,
                           const float* __restrict__ bias,
                           float* __restrict__ out)
{
    __shared__ float pooled[TILE_M * LDS_STRIDE]; // 64 x 128 (+pad) = 33.8 KB
    __shared__ int   nbr[TILE_M * Kc];            // batch-combined neighbor row ids

    const int t   = threadIdx.x;
    const int gp0 = blockIdx.x * TILE_M;          // first global point of this tile

    // ---- Phase 0: stage 64x16 neighbor indices (4 per thread) ----
    #pragma unroll
    for (int j = 0; j < (TILE_M * Kc) / THREADS; ++j) {
        const int q  = t + THREADS * j;           // 0..1023
        const int p  = q >> 4;                    // point in tile
        const int k  = q & 15;                    // neighbor slot
        const int gp = gp0 + p;                   // global point id; b*Nc+n == gp
        const int b  = gp / Nc;
        const long long off = (long long)gp * Kc + k;
        const int idx = knn[off];                 // neighbor index within batch
        nbr[p * Kc + k] = b * Nc + idx;           // combined row id into x
    }
    __syncthreads();

    // ---- Phase 1: gather + mean-pool into LDS (L2-resident gathers) ----
    // 64 points x 16 dim-groups (8 floats) = 1024 tasks -> 4 per thread
    #pragma unroll
    for (int j = 0; j < (TILE_M * (Dc / 8)) / THREADS; ++j) {
        const int q  = t + THREADS * j;           // 0..1023
        const int p  = q & (TILE_M - 1);          // point in tile
        const int g  = q >> 6;                    // dim group (0..15)
        const int d0 = g * 8;
        float acc[8];
        #pragma unroll
        for (int i = 0; i < 8; ++i) acc[i] = 0.f;
        #pragma unroll
        for (int k = 0; k < Kc; ++k) {
            const int row = nbr[p * Kc + k];
            const float4* src = (const float4*)(x + (long long)row * Dc + d0);
            float4 lo = src[0];
            float4 hi = src[1];
            acc[0] += lo.x; acc[1] += lo.y; acc[2] += lo.z; acc[3] += lo.w;
            acc[4] += hi.x; acc[5] += hi.y; acc[6] += hi.z; acc[7] += hi.w;
        }
        const float scale = 1.0f / (float)Kc;
        float* dst = &pooled[p * LDS_STRIDE + d0];
        #pragma unroll
        for (int i = 0; i < 8; ++i) dst[i] = acc[i] * scale;
    }
    __syncthreads();

    // ---- Phase 2: out[64 x 128] = pooled[64 x 128] * W^T via f32 WMMA ----
    // Wave w owns output features e = w*16..w*16+15; 4 M-subtiles share each
    // B fragment (4x reuse of W traffic). 128 WMMAs per wave.
    // A (16x4 f32): lanes 0-15 K=kb,kb+1; lanes 16-31 K=kb+2,kb+3.
    // B (4x16 f32): lane n holds B[k][n] = W[e][d]; same K striping as A.
    const int wave = t >> 5;
    const int lane = t & 31;
    const int col  = lane & 15;
    const int hi   = lane >> 4;
    const int e    = wave * 16 + col;             // output feature
    const int arow = lane & 15;                   // row within an M-subtile

    v8f c[SUBT];
    #pragma unroll
    for (int s = 0; s < SUBT; ++s) c[s] = (v8f){};

    #pragma unroll
    for (int kk = 0; kk < Dc / 4; ++kk) {
        const int kb = kk * 4 + hi * 2;
        const v2f bb = *(const v2f*)(W + e * Dc + kb);
        #pragma unroll
        for (int s = 0; s < SUBT; ++s) {
            const v2f a = *(const v2f*)&pooled[(s * 16 + arow) * LDS_STRIDE + kb];
            c[s] = __builtin_amdgcn_wmma_f32_16x16x4_f32(false, a, false, bb,
                                                         (short)0, c[s], false, false);
        }
    }

    // ---- Epilogue: bias add + store. C/D layout: lane=N, VGPR v -> M=v(+8) ----
    const float bv = bias[e];
    #pragma unroll
    for (int s = 0; s < SUBT; ++s) {
        #pragma unroll
        for (int v = 0; v < 8; ++v) {
            const int m  = s * 16 + v + hi * 8;
            const int pt = gp0 + m;
            out[(long long)pt * Dc + e] = c[s][v] + bv;
        }
    }
}

extern "C" void kernel_launch(void* const* d_in, const int* in_sizes, int n_in,
                              void* d_out, int out_size, void* d_ws, size_t ws_size,
                              hipStream_t stream)
{
    const float* x    = (const float*)d_in[0];
    const int*   knn  = (const int*)d_in[1];
    const float* W    = (const float*)d_in[2];
    const float* bias = (const float*)d_in[3];
    float* out = (float*)d_out;

    const int total_pts = Bc * Nc;                // 200000, multiple of 64
    dim3 grid(total_pts / TILE_M);                // 3125 blocks
    dim3 block(THREADS);                          // 8 wave32s
    mp_pool_linear_kernel<<<grid, block, 0, stream>>>(x, knn, W, bias, out);
}